// SSNet_51307679318697
// MI455X (gfx1250) — compile-verified
//
#include <hip/hip_runtime.h>
#include <hip/hip_bf16.h>

// fp32 WMMA fragments for V_WMMA_F32_16X16X4_F32 (wave32):
//   A: 16x4  -> 2 VGPRs/lane (lane l<16: row l, K={k,k+1}; lane l+16: row l, K={k+2,k+3})
//   B: 4x16  -> 2 VGPRs/lane (lane l<16: col l, K rows {k,k+1}; lanes 16..31: K rows {k+2,k+3})
//   C/D: 16x16 -> 8 VGPRs (VGPR r: lanes 0-15 = row r, lanes 16-31 = row r+8)
typedef __attribute__((ext_vector_type(2))) float v2f;
typedef __attribute__((ext_vector_type(8))) float v8f;

#define BN_EPS 1e-5f
#define NEG_INF (-__builtin_huge_valf())

// ---------------- layer1: conv 3->1 k3 valid + BN + ReLU (Cout=1: VALU) ----------------
__global__ void k_conv1_bn_relu(const float* __restrict__ x, const float* __restrict__ w,
                                const float* __restrict__ bias,
                                const float* __restrict__ g, const float* __restrict__ be,
                                const float* __restrict__ mu, const float* __restrict__ va,
                                float* __restrict__ out) {
  const int OH = 252, OW = 252, H = 254, W = 254;
  int idx = blockIdx.x * blockDim.x + threadIdx.x;
  if (idx >= 8 * OH * OW) return;
  int ox = idx % OW; int t = idx / OW; int oy = t % OH; int b = t / OH;
  float acc = bias[0];
#pragma unroll
  for (int ci = 0; ci < 3; ++ci)
#pragma unroll
    for (int ky = 0; ky < 3; ++ky)
#pragma unroll
      for (int kx = 0; kx < 3; ++kx)
        acc += x[((b * 3 + ci) * H + oy + ky) * W + ox + kx] * w[(ci * 3 + ky) * 3 + kx];
  float sc = g[0] * rsqrtf(va[0] + BN_EPS);
  acc = acc * sc + (be[0] - mu[0] * sc);
  out[idx] = fmaxf(acc, 0.0f);
}

// ---------------- per-ROI adaptive max pool 62x62, batch-flipped, into concat ch 45..49 ----
__global__ void k_roipool(const float* __restrict__ roiX, const int* __restrict__ ROI,
                          float* __restrict__ concat) {
  const int OUTS = 62, HP = 252, B = 8, R = 5, CC = 50;
  int idx = blockIdx.x * blockDim.x + threadIdx.x;
  if (idx >= B * R * OUTS * OUTS) return;
  int j = idx % OUTS; int t = idx / OUTS;
  int i = t % OUTS; t /= OUTS;
  int r = t % R; int bo = t / R;
  int bs = B - 1 - bo;  // jnp.flip(feats, axis=0)
  const int* box = ROI + (bs * R + r) * 4;
  int y1 = (int)((float)box[0] * 0.9f);
  int y2 = (int)((float)box[1] * 0.9f);
  int x1 = (int)((float)box[2] * 0.9f);
  int x2 = (int)((float)box[3] * 0.9f);
  int hh = y2 - y1, ww = x2 - x1;
  int rs = y1 + (i * hh) / OUTS;
  int re = y1 + ((i + 1) * hh + OUTS - 1) / OUTS;
  int cs = x1 + (j * ww) / OUTS;
  int ce = x1 + ((j + 1) * ww + OUTS - 1) / OUTS;
  float mval = NEG_INF;
  for (int yy = rs; yy < re; ++yy)
    for (int xx = cs; xx < ce; ++xx)
      mval = fmaxf(mval, roiX[((size_t)bs * HP + yy) * HP + xx]);
  concat[(((size_t)bo * CC + 45 + r) * OUTS + i) * OUTS + j] = mval;
}

// ---------------- generic implicit-GEMM conv, fp32 WMMA 16x16x4 ----------------
// C[M=OH*OW x N=Cout] = im2col(in) x W^T, fused bias + inference-BN + optional residual + ReLU.
// KS/STRIDE/PAD compile-time: kq->(ky,kx) decode and K-padding masks constant-fold after
// unrolling; PAD==0 (valid conv) needs no bounds checks at all. Out-of-tile A rows / B cols
// read from clamped (safe) addresses and are simply never stored. Output spatial is flat, so
// the epilogue stores through one base pointer at offset mr (no divisions, no 64-bit muls).
template <int KS, int STRIDE, int PAD>
__global__ void k_conv_wmma(const float* __restrict__ in, const float* __restrict__ w,
                            const float* __restrict__ bias,
                            const float* __restrict__ bng, const float* __restrict__ bnb,
                            const float* __restrict__ bnm, const float* __restrict__ bnv,
                            const float* __restrict__ residual,
                            float* __restrict__ out,
                            int Cin, int H, int W, int Cout, int OH, int OW, int relu) {
  constexpr int KS2 = KS * KS;
  constexpr int STEPS = (KS2 + 3) / 4;  // padded k-steps per input channel
  const int lane = threadIdx.x & 31;
  const int wave = threadIdx.x >> 5;
  const int half = lane >> 4;
  const int l = lane & 15;
  const int M = OH * OW;
  const int mBase = (blockIdx.x * 4 + wave) * 16;
  if (mBase >= M) return;
  const int nBase = blockIdx.y * 16;
  const int b = blockIdx.z;
  const int m = mBase + l;
  const int n = nBase + l;
  const bool mv = m < M;
  const bool nv = n < Cout;
  int oy = 0, ox = 0;
  if (mv) { oy = m / OW; ox = m - oy * OW; }
  const int oyS = oy * STRIDE - PAD;
  const int oxS = ox * STRIDE - PAD;
  const float* __restrict__ inb = in + (size_t)b * Cin * H * W;
  const float* __restrict__ wn = w + (size_t)(nv ? n : 0) * (Cin * KS2);  // OIHW
  v8f acc = {};
  for (int ci = 0; ci < Cin; ++ci) {
    const float* __restrict__ inc = inb + (size_t)ci * H * W;
    const float* __restrict__ wc = wn + ci * KS2;
#pragma unroll
    for (int s = 0; s < STEPS; ++s) {
      float av[2], bv[2];
#pragma unroll
      for (int u = 0; u < 2; ++u) {
        // lane-half-dependent flat k index; both candidates decoded at compile time
        constexpr auto dec = [](int kq) { return kq < KS * KS ? kq : 0; };
        const int kq0 = dec(s * 4 + u);       // half == 0
        const int kq1 = dec(s * 4 + 2 + u);   // half == 1
        const int ky0 = kq0 / KS, kx0 = kq0 % KS;
        const int ky1 = kq1 / KS, kx1 = kq1 % KS;
        const bool ok0 = (s * 4 + u) < KS2;
        const bool ok1 = (s * 4 + 2 + u) < KS2;
        const int ky = half ? ky1 : ky0;
        const int kx = half ? kx1 : kx0;
        const bool okk = half ? ok1 : ok0;    // K-padding mask (mostly constant-true)
        const int iy = oyS + ky;
        const int ix = oxS + kx;
        bool inr = true;
        if (PAD != 0) inr = (iy >= 0) && (iy < H) && (ix >= 0) && (ix < W);
        float a = 0.f;
        if (okk && inr) a = inc[iy * W + ix];
        float bwt = 0.f;
        if (okk) bwt = wc[s * 4 + half * 2 + u];
        av[u] = a;
        bv[u] = bwt;
      }
      v2f Af = {av[0], av[1]};
      v2f Bf = {bv[0], bv[1]};
      acc = __builtin_amdgcn_wmma_f32_16x16x4_f32(false, Af, false, Bf, (short)0, acc,
                                                  false, false);
    }
  }
  if (!nv) return;
  float sc = 1.f, sh = 0.f;
  if (bng) { float s = bng[n] * rsqrtf(bnv[n] + BN_EPS); sc = s; sh = bnb[n] - bnm[n] * s; }
  float bi = bias ? bias[n] : 0.f;
  // flat output/residual indexing: element (mr, n) lives at plane_base + mr
  const size_t plane = (size_t)(b * Cout + n) * M;
  float* __restrict__ obase = out + plane + mBase + half * 8;
  const float* __restrict__ rbase = residual ? residual + plane + mBase + half * 8 : nullptr;
  const int mrem = M - (mBase + half * 8);  // rows r with r < mrem are in range
#pragma unroll
  for (int r = 0; r < 8; ++r) {
    if (r >= mrem) continue;
    float val = (acc[r] + bi) * sc + sh;
    if (rbase) val += rbase[r];
    if (relu) val = fmaxf(val, 0.f);
    obase[r] = val;
  }
}

// ---------------- generic max pool (writes into a buffer with outCtot channels at cOff) ----
__global__ void k_maxpool(const float* __restrict__ in, float* __restrict__ out,
                          int B, int C, int H, int W, int OH, int OW,
                          int ks, int stride, int pad, int outCtot, int cOff) {
  int idx = blockIdx.x * blockDim.x + threadIdx.x;
  if (idx >= B * C * OH * OW) return;
  int ox = idx % OW; int t = idx / OW;
  int oy = t % OH; t /= OH;
  int c = t % C; int b = t / C;
  float mval = NEG_INF;
  for (int ky = 0; ky < ks; ++ky) {
    int iy = oy * stride - pad + ky;
    if (iy < 0 || iy >= H) continue;
    for (int kx = 0; kx < ks; ++kx) {
      int ix = ox * stride - pad + kx;
      if (ix < 0 || ix >= W) continue;
      mval = fmaxf(mval, in[(((size_t)b * C + c) * H + iy) * W + ix]);
    }
  }
  out[(((size_t)b * outCtot + c + cOff) * OH + oy) * OW + ox] = mval;
}

// ---------------- FC: partial[ksplit x M x N] = A[MxK] x W[NxK]^T, fp32 WMMA, K split ------
// kchunk and K must be multiples of 16. Out-of-range rows/cols use clamped pointers and are
// masked only at the store. Weights are a pure one-shot stream (268 MB for fc1, > 192 MB L2)
// -> non-temporal loads + speculative prefetch-ahead to keep the 23.3 TB/s HBM pipe full.
__global__ void k_fc_wmma(const float* __restrict__ A, const float* __restrict__ Wt,
                          float* __restrict__ partial,
                          int Mrows, int N, int K, int kchunk) {
  const int lane = threadIdx.x & 31;
  const int wave = threadIdx.x >> 5;
  const int half = lane >> 4;
  const int l = lane & 15;
  const int nBase = (blockIdx.x * 4 + wave) * 16;
  if (nBase >= N) return;
  const int n = nBase + l;
  const bool nv = n < N;
  const int k0 = blockIdx.y * kchunk;
  const int k1 = min(k0 + kchunk, K);
  const float* __restrict__ wrow = Wt + (size_t)(nv ? n : 0) * K;
  const float* __restrict__ arow = A + (size_t)(l < Mrows ? l : 0) * K;
  v8f acc = {};
  for (int k = k0; k < k1; k += 16) {
    __builtin_prefetch(wrow + k + 1024, 0, 0);  // ~4 KB ahead on the weight stream
#pragma unroll
    for (int s = 0; s < 4; ++s) {
      const int kk = k + s * 4 + half * 2;
      v2f Af = *(const v2f*)(arow + kk);
      v2f Bf = __builtin_nontemporal_load((const v2f*)(wrow + kk));
      acc = __builtin_amdgcn_wmma_f32_16x16x4_f32(false, Af, false, Bf, (short)0, acc,
                                                  false, false);
    }
  }
  if (!nv) return;
  // rows r of this half-lane map to output rows half*8 + r; store via one base pointer
  float* __restrict__ pbase =
      partial + ((size_t)blockIdx.y * Mrows + half * 8) * N + n;
  const int mrem = Mrows - half * 8;
#pragma unroll
  for (int r = 0; r < 8; ++r) {
    if (r >= mrem) continue;
    pbase[(size_t)r * N] = acc[r];
  }
}

// deterministic K-split reduction + bias + ReLU
__global__ void k_fc_reduce(const float* __restrict__ partial, const float* __restrict__ bias,
                            float* __restrict__ out, int Mrows, int N, int ksplit, int relu) {
  int idx = blockIdx.x * blockDim.x + threadIdx.x;
  if (idx >= Mrows * N) return;
  int n = idx % N, m = idx / N;
  float s = bias[n];
  for (int p = 0; p < ksplit; ++p) s += partial[((size_t)p * Mrows + m) * N + n];
  if (relu) s = fmaxf(s, 0.f);
  out[(size_t)m * N + n] = s;
}

// ---------------- fc3 (2048->2) + softmax: tiny, VALU ----------------
__global__ void k_fc3_softmax(const float* __restrict__ h, const float* __restrict__ w,
                              const float* __restrict__ bias, float* __restrict__ out) {
  __shared__ float logits[16];
  int t = threadIdx.x;
  int m = (t >> 1) & 7, n = t & 1;
  float acc = bias[n];
  for (int k = 0; k < 2048; ++k) acc += h[(size_t)m * 2048 + k] * w[(size_t)n * 2048 + k];
  if (t < 16) logits[t] = acc;
  __syncthreads();
  if (t < 16) {
    float a = logits[m * 2], b2 = logits[m * 2 + 1];
    float mx = fmaxf(a, b2);
    float ea = __expf(a - mx), eb = __expf(b2 - mx);
    out[m * 2 + n] = (n == 0 ? ea : eb) / (ea + eb);
  }
}

extern "C" void kernel_launch(void* const* d_in, const int* in_sizes, int n_in,
                              void* d_out, int out_size, void* d_ws, size_t ws_size,
                              hipStream_t stream) {
  // -------- flattened inputs (setup_inputs dict order, tuples/dicts depth-first) --------
  const float* x    = (const float*)d_in[0];   // (8,3,254,254)
  const int*   ROI  = (const int*)d_in[1];     // (8,5,4)
  const float* c1w  = (const float*)d_in[3];
  const float* c1b  = (const float*)d_in[4];
  const float* bn1g = (const float*)d_in[5],  *bn1b = (const float*)d_in[6];
  const float* bn1m = (const float*)d_in[7],  *bn1v = (const float*)d_in[8];
  const float* c2w  = (const float*)d_in[9];
  const float* bn2g = (const float*)d_in[10], *bn2b = (const float*)d_in[11];
  const float* bn2m = (const float*)d_in[12], *bn2v = (const float*)d_in[13];
  const float* c3w  = (const float*)d_in[14];
  const float* c3b  = (const float*)d_in[15];
  const float* bn3g = (const float*)d_in[16], *bn3b = (const float*)d_in[17];
  const float* bn3m = (const float*)d_in[18], *bn3v = (const float*)d_in[19];
  const float* bc1w = (const float*)d_in[20];
  const float* bb1g = (const float*)d_in[21], *bb1b = (const float*)d_in[22];
  const float* bb1m = (const float*)d_in[23], *bb1v = (const float*)d_in[24];
  const float* bc2w = (const float*)d_in[25];
  const float* bb2g = (const float*)d_in[26], *bb2b = (const float*)d_in[27];
  const float* bb2m = (const float*)d_in[28], *bb2v = (const float*)d_in[29];
  const float* cxw  = (const float*)d_in[30];
  const float* cxb  = (const float*)d_in[31];
  const float* bnxg = (const float*)d_in[32], *bnxb = (const float*)d_in[33];
  const float* bnxm = (const float*)d_in[34], *bnxv = (const float*)d_in[35];
  const float* fc1w = (const float*)d_in[36], *fc1b = (const float*)d_in[37];
  const float* fc2w = (const float*)d_in[38], *fc2b = (const float*)d_in[39];
  const float* fc3w = (const float*)d_in[40], *fc3b = (const float*)d_in[41];
  float* out = (float*)d_out;

  // -------- workspace layout (fp32), ~50 MB total --------
  float* wsf = (float*)d_ws;
  size_t off = 0;
  auto alloc = [&](size_t nelem) { float* p = wsf + off; off += (nelem + 63) & ~(size_t)63; return p; };
  float* roiX   = alloc((size_t)8 * 252 * 252);        // layer1 output
  float* concat = alloc((size_t)8 * 50 * 62 * 62);     // [norX(45) | roi feats(5)]
  float* c2o    = alloc((size_t)8 * 45 * 124 * 124);   // conv2 pre-pool
  float* h3pre  = alloc((size_t)8 * 64 * 62 * 62);
  float* h3     = alloc((size_t)8 * 64 * 31 * 31);
  float* bt1    = alloc((size_t)8 * 64 * 31 * 31);
  float* bt2    = alloc((size_t)8 * 64 * 31 * 31);
  float* hxpre  = alloc((size_t)8 * 128 * 31 * 31);
  float* hx     = alloc((size_t)8 * 128 * 16 * 16);    // == flattened (8, 32768)
  float* fc1p   = alloc((size_t)32 * 8 * 2048);        // fc1 K-split partials
  float* fc1o   = alloc((size_t)8 * 2048);
  float* fc2p   = alloc((size_t)4 * 8 * 2048);
  float* fc2o   = alloc((size_t)8 * 2048);
  (void)ws_size; (void)in_sizes; (void)n_in; (void)out_size;

  const int TPB = 256;
  auto blks = [](long long n, int t) { return (unsigned)((n + t - 1) / t); };

  // 1) layer1 conv+BN+ReLU
  k_conv1_bn_relu<<<blks(8LL * 252 * 252, TPB), TPB, 0, stream>>>(
      x, c1w, c1b, bn1g, bn1b, bn1m, bn1v, roiX);
  // 2) ROI adaptive max pool (flipped batch) -> concat channels 45..49
  k_roipool<<<blks(8LL * 5 * 62 * 62, TPB), TPB, 0, stream>>>(roiX, ROI, concat);
  // 3) conv2 7x7 s2 valid (no bias) + BN + ReLU  (M=15376, N=45, K=147) - no bounds checks
  {
    dim3 g((961 + 3) / 4, 3, 8);
    k_conv_wmma<7, 2, 0><<<g, 128, 0, stream>>>(x, c2w, nullptr, bn2g, bn2b, bn2m, bn2v,
                                                nullptr, c2o, 3, 254, 254, 45, 124, 124, 1);
  }
  // 4) maxpool 2x2 s2 -> concat channels 0..44
  k_maxpool<<<blks(8LL * 45 * 62 * 62, TPB), TPB, 0, stream>>>(
      c2o, concat, 8, 45, 124, 124, 62, 62, 2, 2, 0, 50, 0);
  // 5) conv3 3x3 p1 (50->64) + bias + BN + ReLU  (M=3844, N=64, K=450)
  {
    dim3 g((241 + 3) / 4, 4, 8);
    k_conv_wmma<3, 1, 1><<<g, 128, 0, stream>>>(concat, c3w, c3b, bn3g, bn3b, bn3m, bn3v,
                                                nullptr, h3pre, 50, 62, 62, 64, 62, 62, 1);
  }
  // 6) maxpool 3x3 s2 p1 : 62 -> 31
  k_maxpool<<<blks(8LL * 64 * 31 * 31, TPB), TPB, 0, stream>>>(
      h3pre, h3, 8, 64, 62, 62, 31, 31, 3, 2, 1, 64, 0);
  // 7) BasicBlock conv1 + BN + ReLU  (M=961, N=64, K=576)
  {
    dim3 g((61 + 3) / 4, 4, 8);
    k_conv_wmma<3, 1, 1><<<g, 128, 0, stream>>>(h3, bc1w, nullptr, bb1g, bb1b, bb1m, bb1v,
                                                nullptr, bt1, 64, 31, 31, 64, 31, 31, 1);
    // 8) BasicBlock conv2 + BN + residual(h3) + ReLU
    k_conv_wmma<3, 1, 1><<<g, 128, 0, stream>>>(bt1, bc2w, nullptr, bb2g, bb2b, bb2m, bb2v,
                                                h3, bt2, 64, 31, 31, 64, 31, 31, 1);
    // 9) layer_x conv 3x3 p1 (64->128) + bias + BN + ReLU  (N=128)
    dim3 gx((61 + 3) / 4, 8, 8);
    k_conv_wmma<3, 1, 1><<<gx, 128, 0, stream>>>(bt2, cxw, cxb, bnxg, bnxb, bnxm, bnxv,
                                                 nullptr, hxpre, 64, 31, 31, 128, 31, 31, 1);
  }
  // 10) maxpool 3x3 s2 p1 : 31 -> 16 ; result is the flattened FC input (NCHW order)
  k_maxpool<<<blks(8LL * 128 * 16 * 16, TPB), TPB, 0, stream>>>(
      hxpre, hx, 8, 128, 31, 31, 16, 16, 3, 2, 1, 128, 0);
  // 11) fc1: M=8, N=2048, K=32768, ksplit=32 (saturate HBM on the 268 MB weight stream)
  {
    dim3 g((128 + 3) / 4, 32, 1);
    k_fc_wmma<<<g, 128, 0, stream>>>(hx, fc1w, fc1p, 8, 2048, 32768, 1024);
    k_fc_reduce<<<blks(8LL * 2048, TPB), TPB, 0, stream>>>(fc1p, fc1b, fc1o, 8, 2048, 32, 1);
  }
  // 12) fc2: M=8, N=2048, K=2048, ksplit=4
  {
    dim3 g((128 + 3) / 4, 4, 1);
    k_fc_wmma<<<g, 128, 0, stream>>>(fc1o, fc2w, fc2p, 8, 2048, 2048, 512);
    k_fc_reduce<<<blks(8LL * 2048, TPB), TPB, 0, stream>>>(fc2p, fc2b, fc2o, 8, 2048, 4, 1);
  }
  // 13) fc3 + softmax -> d_out (8,2)
  k_fc3_softmax<<<1, 32, 0, stream>>>(fc2o, fc3w, fc3b, out);
}